// GCN_20315195310786
// MI455X (gfx1250) — compile-verified
//
#include <hip/hip_runtime.h>
#include <hip/hip_bf16.h>

#define N_NODES  100000
#define N_EDGES  1600000
#define N_GRAPHS 64

typedef __attribute__((ext_vector_type(2))) float v2f;
typedef __attribute__((ext_vector_type(8))) float v8f;

// ---------------------------------------------------------------- utilities
__global__ void fill_zero(float* __restrict__ p, int n) {
  int i = blockIdx.x * blockDim.x + threadIdx.x;
  if (i < n) p[i] = 0.0f;
}

__global__ void degree_kernel(const int* __restrict__ src, const int* __restrict__ dst,
                              float* __restrict__ deg_out, float* __restrict__ deg_in) {
  int e = blockIdx.x * blockDim.x + threadIdx.x;
  if (e < N_EDGES) {
    atomicAdd(&deg_out[src[e]], 1.0f);
    atomicAdd(&deg_in[dst[e]], 1.0f);
  }
}

__global__ void inv_sqrt_kernel(float* __restrict__ deg, int n) {
  int i = blockIdx.x * blockDim.x + threadIdx.x;
  if (i < n) deg[i] = 1.0f / sqrtf(fmaxf(deg[i], 1.0f));
}

// ------------------------------------------------- edge gather + scatter-add
// agg[dst] += x[src] * inv_sqrt_out[src]; one thread per (edge, 4-feature chunk)
template <int D>
__global__ void scatter_kernel(const float* __restrict__ x, const int* __restrict__ src,
                               const int* __restrict__ dst, const float* __restrict__ iso,
                               float* __restrict__ agg) {
  constexpr int C = D / 4;
  int t = blockIdx.x * blockDim.x + threadIdx.x;
  int e = t / C;
  int c = t % C;
  if (e >= N_EDGES) return;
  int s = src[e], d = dst[e];
  float sc = iso[s];
  float4 v = ((const float4*)(x + (size_t)s * D))[c];
  float* a = agg + (size_t)d * D + c * 4;
  atomicAdd(a + 0, v.x * sc);
  atomicAdd(a + 1, v.y * sc);
  atomicAdd(a + 2, v.z * sc);
  atomicAdd(a + 3, v.w * sc);
}

// ---------------------------------------- WMMA fp32 node-feature transform
// out[16-node tile] = relu( (agg * inv_sqrt_in[:,None]) @ W + b )
// Uses V_WMMA_F32_16X16X4_F32: A 16x4 f32 (2 VGPRs), B 4x16 f32 (2 VGPRs),
// C/D 16x16 f32 (8 VGPRs). One wave per (node-tile, 16-col tile).
template <int DIN, int DOUT, bool RELU>
__global__ void transform_wmma(const float* __restrict__ agg, const float* __restrict__ isi,
                               const float* __restrict__ W, const float* __restrict__ bias,
                               float* __restrict__ out) {
  constexpr int NT   = (DOUT + 15) / 16;   // column tiles
  constexpr int OSTR = NT * 16;            // padded output stride
  constexpr int MT   = N_NODES / 16;       // 6250 exact node tiles

  int wave  = (blockIdx.x * blockDim.x + threadIdx.x) >> 5;
  int lane  = threadIdx.x & 31;
  int mtile = wave / NT;
  int ntile = wave % NT;
  if (mtile >= MT) return;                 // wave-uniform exit: EXEC stays all-1s

  int hi   = lane >> 4;                    // 0: K=k,k+1   1: K=k+2,k+3 (A) / same for B
  int lo   = lane & 15;                    // M index (A) / N index (B, C, D)
  int node = mtile * 16 + lo;
  int ncol = ntile * 16 + lo;
  float scale = isi[node];                 // D_in^{-1/2} row scale folded into A
  const float* arow = agg + (size_t)node * DIN;

  v8f c = {0.f, 0.f, 0.f, 0.f, 0.f, 0.f, 0.f, 0.f};
#pragma unroll
  for (int kc = 0; kc < DIN; kc += 4) {
    int k0 = kc + hi * 2;
    v2f a, b;
    a.x = arow[k0] * scale;
    a.y = arow[k0 + 1] * scale;
    bool ok = (ncol < DOUT);               // constant-folds away except layer 3
    b.x = ok ? W[k0 * DOUT + ncol] : 0.0f;
    b.y = ok ? W[(k0 + 1) * DOUT + ncol] : 0.0f;
    c = __builtin_amdgcn_wmma_f32_16x16x4_f32(false, a, false, b, (short)0, c,
                                              false, false);
  }

  float bv = (ncol < DOUT) ? bias[ncol] : 0.0f;
#pragma unroll
  for (int i = 0; i < 8; ++i) {
    int row = mtile * 16 + hi * 8 + i;     // C/D layout: VGPR i holds M=i (+8 for hi lanes)
    float v = c[i] + bv;
    if (RELU) v = fmaxf(v, 0.0f);
    out[(size_t)row * OSTR + ntile * 16 + lo] = v;
  }
}

// --------------------------------------------------------- per-graph pooling
// graph_ids are sorted -> waves are almost always uniform; reduce in-wave first
__global__ void pool_accum(const float* __restrict__ h, const int* __restrict__ gid,
                           float* __restrict__ sums, float* __restrict__ counts) {
  int i = blockIdx.x * blockDim.x + threadIdx.x;
  bool valid = (i < N_NODES);
  int g = valid ? gid[i] : -1;
  float v[10];
#pragma unroll
  for (int j = 0; j < 10; ++j) v[j] = valid ? h[(size_t)i * 16 + j] : 0.0f;

  int g0 = __shfl(g, 0, 32);
  bool uniform = __all(g == g0);
  if (uniform && g0 >= 0) {
#pragma unroll
    for (int j = 0; j < 10; ++j) {
      float t = v[j];
      t += __shfl_xor(t, 16, 32);
      t += __shfl_xor(t, 8, 32);
      t += __shfl_xor(t, 4, 32);
      t += __shfl_xor(t, 2, 32);
      t += __shfl_xor(t, 1, 32);
      v[j] = t;
    }
    if ((threadIdx.x & 31) == 0) {
#pragma unroll
      for (int j = 0; j < 10; ++j) atomicAdd(&sums[g0 * 10 + j], v[j]);
      atomicAdd(&counts[g0], 32.0f);
    }
  } else if (valid) {
#pragma unroll
    for (int j = 0; j < 10; ++j) atomicAdd(&sums[g * 10 + j], v[j]);
    atomicAdd(&counts[g], 1.0f);
  }
}

__global__ void pool_final(const float* __restrict__ sums, const float* __restrict__ counts,
                           float* __restrict__ out) {
  int i = blockIdx.x * blockDim.x + threadIdx.x;
  if (i < N_GRAPHS * 10) out[i] = sums[i] / fmaxf(counts[i / 10], 1.0f);
}

// ------------------------------------------------------------------- driver
extern "C" void kernel_launch(void* const* d_in, const int* in_sizes, int n_in,
                              void* d_out, int out_size, void* d_ws, size_t ws_size,
                              hipStream_t stream) {
  const float* n_feat = (const float*)d_in[0];
  const int*   src    = (const int*)d_in[1];
  const int*   dst    = (const int*)d_in[2];
  const int*   gid    = (const int*)d_in[3];
  const float* W1     = (const float*)d_in[4];
  const float* b1     = (const float*)d_in[5];
  const float* W2     = (const float*)d_in[6];
  const float* b2     = (const float*)d_in[7];
  const float* W3     = (const float*)d_in[8];
  const float* b3     = (const float*)d_in[9];
  float* out = (float*)d_out;

  // workspace layout (floats): [2N degrees/inv-sqrt][32N agg][32N h][640 sums][64 counts]
  float* ws       = (float*)d_ws;
  const int N     = N_NODES;
  float* inv_out  = ws;              // deg_out -> inv_sqrt_out (in place)
  float* inv_in   = ws + N;          // deg_in  -> inv_sqrt_in  (in place)
  float* agg      = ws + 2 * (size_t)N;
  float* h        = ws + 34 * (size_t)N;
  float* sums     = ws + 66 * (size_t)N;
  float* counts   = sums + N_GRAPHS * 10;

  const int B = 256;
  auto cdiv = [](long a, long b) { return (int)((a + b - 1) / b); };

  // degrees -> D^{-1/2}
  fill_zero<<<cdiv(2 * N, B), B, 0, stream>>>(inv_out, 2 * N);
  degree_kernel<<<cdiv(N_EDGES, B), B, 0, stream>>>(src, dst, inv_out, inv_in);
  inv_sqrt_kernel<<<cdiv(2 * N, B), B, 0, stream>>>(inv_out, 2 * N);

  const int MT = N_NODES / 16;  // 6250

  // layer 1: 8 -> 16 (+relu)
  fill_zero<<<cdiv(8L * N, B), B, 0, stream>>>(agg, 8 * N);
  scatter_kernel<8><<<cdiv(2L * N_EDGES, B), B, 0, stream>>>(n_feat, src, dst, inv_out, agg);
  transform_wmma<8, 16, true><<<cdiv(MT * 1L * 32, B), B, 0, stream>>>(agg, inv_in, W1, b1, h);

  // layer 2: 16 -> 32 (+relu)
  fill_zero<<<cdiv(16L * N, B), B, 0, stream>>>(agg, 16 * N);
  scatter_kernel<16><<<cdiv(4L * N_EDGES, B), B, 0, stream>>>(h, src, dst, inv_out, agg);
  transform_wmma<16, 32, true><<<cdiv(MT * 2L * 32, B), B, 0, stream>>>(agg, inv_in, W2, b2, h);

  // layer 3: 32 -> 10 (padded to 16, no relu)
  fill_zero<<<cdiv(32L * N, B), B, 0, stream>>>(agg, 32 * N);
  scatter_kernel<32><<<cdiv(8L * N_EDGES, B), B, 0, stream>>>(h, src, dst, inv_out, agg);
  transform_wmma<32, 10, false><<<cdiv(MT * 1L * 32, B), B, 0, stream>>>(agg, inv_in, W3, b3, h);

  // per-graph mean
  fill_zero<<<cdiv(N_GRAPHS * 11, B), B, 0, stream>>>(sums, N_GRAPHS * 11);
  pool_accum<<<cdiv(N, B), B, 0, stream>>>(h, gid, sums, counts);
  pool_final<<<cdiv(N_GRAPHS * 10, B), B, 0, stream>>>(sums, counts, out);
}